// NeRF_54116587930011
// MI455X (gfx1250) — compile-verified
//
#include <hip/hip_runtime.h>
#include <stdint.h>

// NeRF volumetric alpha-compositing render for MI455X (gfx1250, wave32).
// Memory-bound streaming kernel (roofline: ~335 MB @ 23.3 TB/s ≈ 14.4 us).
// One wave per ray; async global->LDS triple-buffered rgbo loads (ASYNCcnt);
// wave32 shuffle scan for log-space transmittance; v_exp/v_rcp sigmoid.

#define NERF_N_RAYS    131072
#define NERF_N_SAMPLES 128
#define NERF_FAR_DELTA 1e9f

#define WAVES_PER_BLOCK 8
#define CHUNK           32                       // samples per chunk == wave32
#define NCHUNK          (NERF_N_SAMPLES / CHUNK) // 4
#define NBUF            3                        // triple buffer: no LDS WAR hazard

__global__ __launch_bounds__(WAVES_PER_BLOCK * 32)
void nerf_render_kernel(const float* __restrict__ rgbo,   // [N, S, 4]
                        const float* __restrict__ depth,  // [N, S]
                        float* __restrict__ out)          // [N, 3]
{
    // Per-wave triple buffer: 3 chunks * 32 lanes * float4 = 1.5 KB/wave, 12 KB/block.
    __shared__ float4 sbuf[WAVES_PER_BLOCK * NBUF * CHUNK];

    const int lane = threadIdx.x & 31;
    const int wave = threadIdx.x >> 5;
    const int ray  = blockIdx.x * WAVES_PER_BLOCK + wave;

    float4* mybuf = &sbuf[wave * NBUF * CHUNK];
    const uint64_t rgbo_base = (uint64_t)(uintptr_t)rgbo;

    // Async-copy one chunk of rgbo (16 B/lane, 512 B/wave, fully coalesced)
    // into LDS buffer `buf`. Tracked by ASYNCcnt, not LOADcnt.
    auto issue_chunk = [&](int chunk, int buf) {
        // Low 32 bits of a generic shared pointer == workgroup-relative LDS
        // byte offset (LDS aperture uses the low 32 address bits); hardware adds LDS_BASE.
        uint32_t lds_addr = (uint32_t)(uintptr_t)&mybuf[buf * CHUNK + lane];
        uint32_t voff = ((uint32_t)ray * (NERF_N_SAMPLES * 16u))
                      + (uint32_t)(chunk * CHUNK + lane) * 16u;
        asm volatile("global_load_async_to_lds_b128 %0, %1, %2"
                     :: "v"(lds_addr), "v"(voff), "s"(rgbo_base)
                     : "memory");
    };

    issue_chunk(0, 0);

    const float* dptr = depth + (size_t)ray * NERF_N_SAMPLES;

    float carry = 0.0f;                 // running sum of mult (log transmittance)
    float ar = 0.0f, ag = 0.0f, ab = 0.0f;

#pragma unroll
    for (int c = 0; c < NCHUNK; ++c) {
        if (c + 1 < NCHUNK) {
            issue_chunk(c + 1, (c + 1) % NBUF);
            // Let the just-issued prefetch stay in flight; wait for chunk c only.
            asm volatile("s_wait_asynccnt 1" ::: "memory");
        } else {
            asm volatile("s_wait_asynccnt 0" ::: "memory");
        }

        // Buffer (c % NBUF) is not overwritten again until the prefetch issued
        // at iteration c+2, i.e. well after this read's data is consumed.
        float4 v = mybuf[(c % NBUF) * CHUNK + lane];   // ds_load_b128

        int s = c * CHUNK + lane;
        float d0 = dptr[s];
        float delta = (s < NERF_N_SAMPLES - 1) ? (dptr[s + 1] - d0)
                                               : NERF_FAR_DELTA;
        float mult = -v.w * delta;                  // -sigma * delta  (<= 0)

        // Wave32 inclusive scan of mult across lanes (5 shuffle steps).
        float scan = mult;
#pragma unroll
        for (int off = 1; off < 32; off <<= 1) {
            float n = __shfl_up(scan, (unsigned)off, 32);
            if (lane >= off) scan += n;
        }
        float excl  = scan - mult;                  // exclusive prefix within chunk
        float T     = __expf(carry + excl);         // transmittance
        float alpha = 1.0f - __expf(mult);
        float w     = T * alpha;

        // sigmoid(x) = 1 / (1 + exp(-x)) via v_exp_f32 + v_rcp_f32
        ar += w * __builtin_amdgcn_rcpf(1.0f + __expf(-v.x));
        ag += w * __builtin_amdgcn_rcpf(1.0f + __expf(-v.y));
        ab += w * __builtin_amdgcn_rcpf(1.0f + __expf(-v.z));

        carry += __shfl(scan, 31, 32);              // chunk total -> carry
    }

    // Butterfly reduction of the three accumulators across the wave.
#pragma unroll
    for (int off = 16; off > 0; off >>= 1) {
        ar += __shfl_xor(ar, off, 32);
        ag += __shfl_xor(ag, off, 32);
        ab += __shfl_xor(ab, off, 32);
    }

    if (lane == 0) {
        float* o = out + (size_t)ray * 3;
        o[0] = ar;
        o[1] = ag;
        o[2] = ab;
    }
}

extern "C" void kernel_launch(void* const* d_in, const int* in_sizes, int n_in,
                              void* d_out, int out_size, void* d_ws, size_t ws_size,
                              hipStream_t stream) {
    (void)in_sizes; (void)n_in; (void)d_ws; (void)ws_size; (void)out_size;
    const float* rgbo  = (const float*)d_in[0];   // [N, S, 4] f32
    const float* depth = (const float*)d_in[1];   // [N, S]    f32
    float* out = (float*)d_out;                   // [N, 3]    f32

    dim3 block(WAVES_PER_BLOCK * 32);
    dim3 grid(NERF_N_RAYS / WAVES_PER_BLOCK);     // 16384 blocks, 1 wave per ray
    nerf_render_kernel<<<grid, block, 0, stream>>>(rgbo, depth, out);
}